// TripletAllLoss_1580547966983
// MI455X (gfx1250) — compile-verified
//
#include <hip/hip_runtime.h>
#include <hip/hip_bf16.h>

typedef float v2f __attribute__((ext_vector_type(2)));
typedef float v8f __attribute__((ext_vector_type(8)));

#define N_EMB 512
#define D_EMB 128
#define EPS_F 1e-4f
#define MARGIN 1.0f

// ---------------------------------------------------------------------------
// Kernel 1: row norms  norms[i] = sum_k A[i,k]^2
// one block (128 threads) per row; coalesced load + LDS tree reduce
// ---------------------------------------------------------------------------
__global__ __launch_bounds__(128) void norms_kernel(const float* __restrict__ A,
                                                    float* __restrict__ norms) {
    const int row = blockIdx.x;
    const int t   = threadIdx.x;
    __shared__ float red[128];
    const float v = A[row * D_EMB + t];
    red[t] = v * v;
    __syncthreads();
    for (int off = 64; off > 0; off >>= 1) {
        if (t < off) red[t] += red[t + off];
        __syncthreads();
    }
    if (t == 0) norms[row] = red[0];
}

// ---------------------------------------------------------------------------
// Kernel 2: dmat via fp32 WMMA.  One wave (32 threads) per 16x16 output tile.
// prod = A * A^T accumulated with v_wmma_f32_16x16x4_f32 over K=128 (32 steps).
// A-matrix layout (16x4 f32): lanes 0-15 hold K={0,1}, lanes 16-31 hold K={2,3},
// each lane's M = lane&15.  B = A^T (4x16): same per-lane load pattern with the
// column-tile row base.  C/D: VGPR v -> M = v + 8*(lane>=16), N = lane&15.
// dmat = sqrt(max(norm_i + norm_j - 2*prod, eps))
// ---------------------------------------------------------------------------
__global__ __launch_bounds__(32) void dmat_wmma_kernel(const float* __restrict__ A,
                                                       const float* __restrict__ norms,
                                                       float* __restrict__ dmat) {
    const int ti   = blockIdx.x;          // row tile
    const int tj   = blockIdx.y;          // col tile
    const int lane = threadIdx.x;         // 0..31
    const int half = lane >> 4;           // 0: K pair {0,1}, 1: K pair {2,3}
    const int l    = lane & 15;

    const float* __restrict__ arow = A + (ti * 16 + l) * D_EMB;
    const float* __restrict__ brow = A + (tj * 16 + l) * D_EMB;

    v8f c = {};
    for (int k = 0; k < D_EMB; k += 4) {
        const int ka = k + 2 * half;
        v2f a = *(const v2f*)(arow + ka);
        v2f b = *(const v2f*)(brow + ka);
        // 8 args: (neg_a, A, neg_b, B, c_mod, C, reuse_a, reuse_b)
        c = __builtin_amdgcn_wmma_f32_16x16x4_f32(false, a, false, b,
                                                  (short)0, c, false, false);
    }

    const int gj = tj * 16 + l;
    const float nj = norms[gj];
#pragma unroll
    for (int v = 0; v < 8; ++v) {
        const int gi = ti * 16 + v + 8 * half;
        float res = norms[gi] + nj - 2.0f * c[v];
        res = fmaxf(res, EPS_F);           // clip(.,0) then clip(.,eps) == max(.,eps)
        dmat[gi * N_EMB + gj] = sqrtf(res);
    }
}

// ---------------------------------------------------------------------------
// Kernel 3: per-anchor triplet partial sums.  One block (256 threads) per
// anchor i.  d[i,:] and labels are LDS-resident.  The k (positive) test is
// block-uniform, so non-positive k cost ~1 compare.  "Negative" j means
// label[j] != label[i] OR j == i  (matches M = pos[:,None,:]*(1-pos[:,:,None])
// with pos diag zeroed).
// partials[2i] = sum relu(d[i,k]-d[i,j]+margin);  partials[2i+1] = count
// ---------------------------------------------------------------------------
__global__ __launch_bounds__(256) void triplet_kernel(const float* __restrict__ dmat,
                                                      const int* __restrict__ labels,
                                                      float* __restrict__ partials) {
    const int i = blockIdx.x;
    const int t = threadIdx.x;

    __shared__ float drow[N_EMB];
    __shared__ int   labs[N_EMB];
    for (int idx = t; idx < N_EMB; idx += 256) {
        drow[idx] = dmat[i * N_EMB + idx];
        labs[idx] = labels[idx];
    }
    __syncthreads();

    const int li = labs[i];
    float num = 0.0f, cnt = 0.0f;

    for (int k = 0; k < N_EMB; ++k) {                 // uniform branch over k
        if ((labs[k] != li) || (k == i)) continue;    // k must be a positive
        const float dk = drow[k] + MARGIN;
        for (int j = t; j < N_EMB; j += 256) {
            const bool pij = (labs[j] == li) && (j != i);
            if (!pij) {                               // j is a valid "negative"
                num += fmaxf(dk - drow[j], 0.0f);
                cnt += 1.0f;
            }
        }
    }

    __shared__ float rn[256];
    __shared__ float rc[256];
    rn[t] = num; rc[t] = cnt;
    __syncthreads();
    for (int off = 128; off > 0; off >>= 1) {
        if (t < off) { rn[t] += rn[t + off]; rc[t] += rc[t + off]; }
        __syncthreads();
    }
    if (t == 0) {
        partials[2 * i]     = rn[0];
        partials[2 * i + 1] = rc[0];
    }
}

// ---------------------------------------------------------------------------
// Kernel 4: deterministic final reduction, out = sum(num) / sum(cnt)
// ---------------------------------------------------------------------------
__global__ __launch_bounds__(256) void finalize_kernel(const float* __restrict__ partials,
                                                       float* __restrict__ out) {
    const int t = threadIdx.x;
    float num = 0.0f, cnt = 0.0f;
    for (int i = t; i < N_EMB; i += 256) {
        num += partials[2 * i];
        cnt += partials[2 * i + 1];
    }
    __shared__ float rn[256];
    __shared__ float rc[256];
    rn[t] = num; rc[t] = cnt;
    __syncthreads();
    for (int off = 128; off > 0; off >>= 1) {
        if (t < off) { rn[t] += rn[t + off]; rc[t] += rc[t + off]; }
        __syncthreads();
    }
    if (t == 0) out[0] = rn[0] / rc[0];
}

// ---------------------------------------------------------------------------
extern "C" void kernel_launch(void* const* d_in, const int* in_sizes, int n_in,
                              void* d_out, int out_size, void* d_ws, size_t ws_size,
                              hipStream_t stream) {
    const float* A      = (const float*)d_in[0];  // [512, 128] fp32
    const int*   labels = (const int*)d_in[1];    // [512] int
    float*       out    = (float*)d_out;          // scalar fp32

    // Workspace layout (all 256B aligned):
    //   [0, 2KB)        norms   (512 f32)
    //   [4KB, 4KB+1MB)  dmat    (512*512 f32)
    //   [+1MB, +1MB+4KB) partials (512*2 f32)
    char* ws = (char*)d_ws;
    float* norms    = (float*)(ws);
    float* dmat     = (float*)(ws + 4096);
    float* partials = (float*)(ws + 4096 + (size_t)N_EMB * N_EMB * sizeof(float));

    norms_kernel<<<N_EMB, 128, 0, stream>>>(A, norms);
    dmat_wmma_kernel<<<dim3(N_EMB / 16, N_EMB / 16), 32, 0, stream>>>(A, norms, dmat);
    triplet_kernel<<<N_EMB, 256, 0, stream>>>(dmat, labels, partials);
    finalize_kernel<<<1, 256, 0, stream>>>(partials, out);
}